// FCNNSlopeValuationFunction_27419071217679
// MI455X (gfx1250) — compile-verified
//
#include <hip/hip_runtime.h>
#include <hip/hip_bf16.h>

// FCNNSlopeValuation zone-pick kernel for MI455X (gfx1250).
//
// Memory-bound: ~400 MB of traffic for B=4M rows -> ~17us at 23.3 TB/s.
// Strategy: stage per-row data into LDS with CDNA5 async global->LDS DMA
// (global_load_async_to_lds_b128/b32, tracked by ASYNCcnt), then do the
// data-dependent dir[zone] gather out of LDS instead of global memory.
// Each wave only consumes LDS written by its own lanes, so a single
// s_wait_asynccnt 0 is the only synchronization required.

#define BLOCK 256
#define ZROW_DW 8   // staged z row stride in dwords (32B; z0..z4 in first 20B)
#define DROW_DW 8   // staged dir row stride in dwords (32B = 8 floats)

__global__ __launch_bounds__(BLOCK)
void FCNNSlopeValuation_zone_pick_kernel(const float* __restrict__ z1,
                                         const float* __restrict__ dirp,
                                         float* __restrict__ out,
                                         int n)
{
    __shared__ float sz[BLOCK * ZROW_DW];   // 8 KB
    __shared__ float sd[BLOCK * DROW_DW];   // 8 KB

    const int t   = threadIdx.x;
    const int row = blockIdx.x * BLOCK + t;
    const int r   = (row < n) ? row : (n - 1);   // clamp tail lanes to a valid row

    // LDS byte offsets: low 32 bits of a flat LDS address are the LDS offset.
    unsigned lz = (unsigned)(size_t)(&sz[t * ZROW_DW]);
    unsigned ld = (unsigned)(size_t)(&sd[t * DROW_DW]);
    // Global byte offsets (B*64 = 256MB and B*32 = 128MB both fit in 32 bits).
    unsigned gz = (unsigned)r * 64u;   // z_1 row stride: 16 floats
    unsigned gd = (unsigned)r * 32u;   // dir row stride:  8 floats

    unsigned long long zb = (unsigned long long)z1;
    unsigned long long db = (unsigned long long)dirp;

    // Async DMA: GVS mode (saddr base + 32-bit vgpr offset). The instruction
    // offset is added to BOTH the LDS and global addresses (ISA 08 §4.4), so
    // the same base registers serve the offset:16 second halves.
    asm volatile(
        "global_load_async_to_lds_b128 %0, %2, %4\n\t"            // z0..z3
        "global_load_async_to_lds_b32  %0, %2, %4 offset:16\n\t"  // z4
        "global_load_async_to_lds_b128 %1, %3, %5\n\t"            // dir0..3
        "global_load_async_to_lds_b128 %1, %3, %5 offset:16\n\t"  // dir4..7
        :
        : "v"(lz), "v"(ld), "v"(gz), "v"(gd), "s"(zb), "s"(db)
        : "memory");

    // Wait for this wave's async transfers to land in LDS.
    asm volatile("s_wait_asynccnt 0x0" ::: "memory");

    const int zbase = t * ZROW_DW;
    const int dbase = t * DROW_DW;

    float has = sz[zbase + 0];           // IDX_LINE
    float lx  = sz[zbase + 1];           // IDX_LX
    float ly  = sz[zbase + 2];           // IDX_LY
    float rx  = sz[zbase + 3];           // IDX_RX
    float ry  = sz[zbase + 4];           // IDX_RY

    float dx = rx - lx;
    float dy = ly - ry;                  // -(ry - ly), screen-coord flip

    // cart2pol angle in degrees mapped to [0, 360)
    float phi = atan2f(dy, dx) * 57.29577951308232f;   // 180/pi
    phi = (phi < 0.0f) ? (phi + 360.0f) : phi;

    // int(phi) truncation (phi >= 0, may round up to exactly 360.0f)
    int iphi  = (int)phi;                // in [0, 360]
    int shift = 90 + iphi;               // in [90, 450]
    if (shift >= 360) shift -= 360;      // == (90 + iphi) % 360

    // area_angle = int(180/8) = 22, half = 11 (exact integers)
    int zone = ((shift + 11) / 22) & 7;  // floor-div then % 8

    float picked = sd[dbase + zone];     // LDS gather

    if (row < n)
        out[row] = (has != 0.0f) ? picked : 0.0f;
}

extern "C" void kernel_launch(void* const* d_in, const int* in_sizes, int n_in,
                              void* d_out, int out_size, void* d_ws, size_t ws_size,
                              hipStream_t stream) {
    (void)in_sizes; (void)n_in; (void)d_ws; (void)ws_size;
    const float* z1  = (const float*)d_in[0];   // [B, 16] float32
    const float* dir = (const float*)d_in[1];   // [B, 8]  float32
    float* out = (float*)d_out;                 // [B]     float32

    const int n = out_size;                     // B rows
    if (n <= 0) return;
    const int blocks = (n + BLOCK - 1) / BLOCK;
    hipLaunchKernelGGL(FCNNSlopeValuation_zone_pick_kernel,
                       dim3(blocks), dim3(BLOCK), 0, stream,
                       z1, dir, out, n);
}